// My_Net2d_14113262535056
// MI455X (gfx1250) — compile-verified
//
#include <hip/hip_runtime.h>
#include <hip/hip_bf16.h>

// ---------------------------------------------------------------------------
// SwiftFormer block for MI455X (gfx1250, wave32).
// All channel-mixing GEMMs use V_WMMA_F32_16X16X4_F32 (fp32-native matrix op).
// ---------------------------------------------------------------------------

typedef __attribute__((ext_vector_type(2))) float v2f;
typedef __attribute__((ext_vector_type(8))) float v8f;

static constexpr int  BATCH = 8;
static constexpr int  CDIM  = 32;
static constexpr int  HSZ   = 256;
static constexpr int  WSZ   = 256;
static constexpr int  HW    = HSZ * WSZ;        // 65536
static constexpr int  OHW   = (HSZ/2)*(WSZ/2);  // 16384
static constexpr int  CE    = 64;
static constexpr long P32   = (long)BATCH * CDIM * HW;   // 16,777,216 floats

__device__ __forceinline__ v8f wmma4(v2f a, v2f b, v8f c) {
  // D = A(16x4 f32) x B(4x16 f32) + C ; emits v_wmma_f32_16x16x4_f32
  return __builtin_amdgcn_wmma_f32_16x16x4_f32(false, a, false, b, (short)0, c,
                                               false, false);
}

__device__ __forceinline__ float gelu_exact(float x) {
  return 0.5f * x * (1.0f + erff(x * 0.70710678118654752440f));
}

// ---------------------------------------------------------------------------
// Depthwise 3x3 (pad 1) + BatchNorm, NCHW streaming kernel. One thread/pixel.
// ---------------------------------------------------------------------------
__global__ __launch_bounds__(256) void dwbn_kernel(
    const float* __restrict__ in, const float* __restrict__ w,
    const float* __restrict__ bias, const float* __restrict__ g,
    const float* __restrict__ be, const float* __restrict__ mu,
    const float* __restrict__ var, float* __restrict__ out) {
  long idx = (long)blockIdx.x * 256 + threadIdx.x;
  int x = (int)(idx & (WSZ - 1));
  int y = (int)((idx >> 8) & (HSZ - 1));
  int c = (int)((idx >> 16) & (CDIM - 1));
  const float* plane = in + (idx & ~(long)(HW - 1));  // base of this (n,c) plane
  float acc = 0.0f;
#pragma unroll
  for (int ty = 0; ty < 3; ++ty) {
    int iy = y + ty - 1;
#pragma unroll
    for (int tx = 0; tx < 3; ++tx) {
      int ix = x + tx - 1;
      bool ok = ((unsigned)iy < (unsigned)HSZ) && ((unsigned)ix < (unsigned)WSZ);
      float v = ok ? plane[iy * WSZ + ix] : 0.0f;
      acc += v * w[c * 9 + ty * 3 + tx];
    }
  }
  float s = g[c] * rsqrtf(var[c] + 1e-5f);
  out[idx] = (acc + bias[c] - mu[c]) * s + be[c];
}

// ---------------------------------------------------------------------------
// Elementwise BatchNorm (NCHW).
// ---------------------------------------------------------------------------
__global__ __launch_bounds__(256) void bn_kernel(
    const float* __restrict__ in, const float* __restrict__ g,
    const float* __restrict__ be, const float* __restrict__ mu,
    const float* __restrict__ var, float* __restrict__ out) {
  long idx = (long)blockIdx.x * 256 + threadIdx.x;
  int c = (int)((idx >> 16) & (CDIM - 1));
  float s = g[c] * rsqrtf(var[c] + 1e-5f);
  out[idx] = (in[idx] - mu[c]) * s + be[c];
}

// ---------------------------------------------------------------------------
// Fused pointwise block: out = res + ls * (W2 @ gelu(W1 @ act + b1) + b2).
// Weights in OIHW ([N][K]). Block = 64 pixels, 8 waves, hidden staged in LDS.
// ---------------------------------------------------------------------------
template <int HID>
__global__ __launch_bounds__(256) void pw_block_kernel(
    const float* __restrict__ res, const float* __restrict__ act,
    const float* __restrict__ W1, const float* __restrict__ b1,
    const float* __restrict__ W2, const float* __restrict__ b2,
    const float* __restrict__ ls, float* __restrict__ out) {
  __shared__ float u[64][HID + 1];
  const int lane  = threadIdx.x & 31;
  const int wave  = threadIdx.x >> 5;
  const int mrow  = lane & 15;
  const int khalf = lane >> 4;
  const long pxbase = (long)blockIdx.x * 64;

  // ---- GEMM1: [64 x 32] @ [32 x HID] -> gelu -> LDS ----
  constexpr int NT1 = HID / 16;
  for (int t = wave; t < 4 * NT1; t += 8) {
    int pxt = t / NT1, nt = t % NT1;
    long px   = pxbase + pxt * 16 + mrow;
    long nimg = px / HW, sp = px % HW;
    const float* abase = act + (nimg * CDIM) * (long)HW + sp;
    int nn = nt * 16 + mrow;  // column index (lane&15)
    v8f acc = {};
#pragma unroll
    for (int ks = 0; ks < CDIM / 4; ++ks) {
      int kb = ks * 4 + khalf * 2;
      v2f a, b;
      a.x = abase[(long)kb * HW];
      a.y = abase[(long)(kb + 1) * HW];
      b.x = W1[nn * CDIM + kb];
      b.y = W1[nn * CDIM + kb + 1];
      acc = wmma4(a, b, acc);
    }
#pragma unroll
    for (int r = 0; r < 8; ++r) {
      int m = r + khalf * 8;
      int n = nt * 16 + mrow;
      u[pxt * 16 + m][n] = gelu_exact(acc[r] + b1[n]);
    }
  }
  __syncthreads();

  // ---- GEMM2: [64 x HID] @ [HID x 32] + residual ----
  {
    int pxt = wave >> 1, nt = wave & 1;
    v8f acc = {};
#pragma unroll
    for (int ks = 0; ks < HID / 4; ++ks) {
      int kb = ks * 4 + khalf * 2;
      v2f a, b;
      a.x = u[pxt * 16 + mrow][kb];
      a.y = u[pxt * 16 + mrow][kb + 1];
      int nn = nt * 16 + mrow;
      b.x = W2[nn * HID + kb];
      b.y = W2[nn * HID + kb + 1];
      acc = wmma4(a, b, acc);
    }
#pragma unroll
    for (int r = 0; r < 8; ++r) {
      int m  = r + khalf * 8;
      int cc = nt * 16 + mrow;
      long p2 = pxbase + pxt * 16 + m;
      long nimg = p2 / HW, sp = p2 % HW;
      long idx = (nimg * CDIM + cc) * (long)HW + sp;
      out[idx] = res[idx] + ls[cc] * (acc[r] + b2[cc]);
    }
  }
}

// ---------------------------------------------------------------------------
// Attention phase 1: q/k projections (WMMA), per-token L2 normalize,
// s = (q . wg) * C^-0.5, block-partial reductions of s^2 and s*q (atomics).
// q/k written token-major [tok][32]. Linear weights stored [K][N].
// ---------------------------------------------------------------------------
__global__ __launch_bounds__(256) void attn_qk_kernel(
    const float* __restrict__ h, const float* __restrict__ Wq,
    const float* __restrict__ bq, const float* __restrict__ Wk,
    const float* __restrict__ bk, const float* __restrict__ wg,
    float* __restrict__ qout, float* __restrict__ kout,
    float* __restrict__ sums) {
  __shared__ float uq[64][33];
  __shared__ float uk[64][33];
  __shared__ float sar[64];
  __shared__ float red[8][32];
  __shared__ float ssqred[8];
  const int lane  = threadIdx.x & 31;
  const int wave  = threadIdx.x >> 5;
  const int mrow  = lane & 15;
  const int khalf = lane >> 4;
  const long pxbase = (long)blockIdx.x * 64;
  const int  bimg   = (int)(pxbase / HW);

  for (int t = wave; t < 16; t += 8) {
    int mat = t & 1, nt = (t >> 1) & 1, pxt = t >> 2;
    long px = pxbase + pxt * 16 + mrow;
    long sp = px % HW;
    const float* abase = h + ((long)bimg * CDIM) * (long)HW + sp;
    const float* W = mat ? Wk : Wq;
    int nn = nt * 16 + mrow;
    v8f acc = {};
#pragma unroll
    for (int ks = 0; ks < 8; ++ks) {
      int kb = ks * 4 + khalf * 2;
      v2f a, b;
      a.x = abase[(long)kb * HW];
      a.y = abase[(long)(kb + 1) * HW];
      b.x = W[kb * CDIM + nn];
      b.y = W[(kb + 1) * CDIM + nn];
      acc = wmma4(a, b, acc);
    }
    const float* bb = mat ? bk : bq;
#pragma unroll
    for (int r = 0; r < 8; ++r) {
      int m = r + khalf * 8;
      int n = nt * 16 + mrow;
      float v = acc[r] + bb[n];
      if (mat) uk[pxt * 16 + m][n] = v;
      else     uq[pxt * 16 + m][n] = v;
    }
  }
  __syncthreads();

  if (threadIdx.x < 64) {
    int tok = threadIdx.x;
    float sq = 0.0f, sk = 0.0f;
#pragma unroll
    for (int c = 0; c < 32; ++c) {
      float a = uq[tok][c]; sq += a * a;
      float b = uk[tok][c]; sk += b * b;
    }
    float rq = 1.0f / fmaxf(sqrtf(sq), 1e-12f);
    float rk = 1.0f / fmaxf(sqrtf(sk), 1e-12f);
    float s = 0.0f;
#pragma unroll
    for (int c = 0; c < 32; ++c) {
      float a = uq[tok][c] * rq;
      uq[tok][c] = a;
      uk[tok][c] *= rk;
      s += a * wg[c];
    }
    sar[tok] = s * 0.17677669529663687f;  // * C^-0.5
  }
  __syncthreads();

  for (int i = threadIdx.x; i < 64 * 32; i += 256) {
    int tok = i >> 5, c = i & 31;
    qout[(pxbase + tok) * 32 + c] = uq[tok][c];
    kout[(pxbase + tok) * 32 + c] = uk[tok][c];
  }
  {
    int g8 = threadIdx.x >> 5, c = threadIdx.x & 31;
    float p = 0.0f;
#pragma unroll
    for (int j = 0; j < 8; ++j) p += sar[g8 * 8 + j] * uq[g8 * 8 + j][c];
    red[g8][c] = p;
    if (c == 0) {
      float q2 = 0.0f;
#pragma unroll
      for (int j = 0; j < 8; ++j) { float s = sar[g8 * 8 + j]; q2 += s * s; }
      ssqred[g8] = q2;
    }
  }
  __syncthreads();
  if (threadIdx.x < 32) {
    float p = 0.0f;
#pragma unroll
    for (int g8 = 0; g8 < 8; ++g8) p += red[g8][threadIdx.x];
    atomicAdd(&sums[bimg * 33 + 1 + threadIdx.x], p);
  } else if (threadIdx.x == 32) {
    float p = 0.0f;
#pragma unroll
    for (int g8 = 0; g8 < 8; ++g8) p += ssqred[g8];
    atomicAdd(&sums[bimg * 33], p);
  }
}

// G[b][c] = (sum_n s_n q_nc) / max(||s||_2, eps)
__global__ void attn_finalize_kernel(const float* __restrict__ sums,
                                     float* __restrict__ G) {
  int b = threadIdx.x >> 5, c = threadIdx.x & 31;
  float nrm = fmaxf(sqrtf(sums[b * 33]), 1e-12f);
  G[b * 32 + c] = sums[b * 33 + 1 + c] / nrm;
}

// ---------------------------------------------------------------------------
// Attention phase 2: o = ((G*k)@Wp + pb + q)@Wf + fb ; h += ls1*o.
// Weights stored [K][N]. q/k token-major.
// ---------------------------------------------------------------------------
__global__ __launch_bounds__(256) void attn_out_kernel(
    const float* __restrict__ res, const float* __restrict__ q,
    const float* __restrict__ k, const float* __restrict__ G,
    const float* __restrict__ Wp, const float* __restrict__ pb,
    const float* __restrict__ Wf, const float* __restrict__ fb,
    const float* __restrict__ ls1, float* __restrict__ out) {
  __shared__ float u[64][33];
  const int lane  = threadIdx.x & 31;
  const int wave  = threadIdx.x >> 5;
  const int mrow  = lane & 15;
  const int khalf = lane >> 4;
  const long pxbase = (long)blockIdx.x * 64;
  const int  bimg   = (int)(pxbase / HW);
  const float* Gb = G + bimg * 32;
  const int pxt = wave >> 1, nt = wave & 1;

  {
    long px = pxbase + pxt * 16 + mrow;
    v8f acc = {};
#pragma unroll
    for (int ks = 0; ks < 8; ++ks) {
      int kb = ks * 4 + khalf * 2;
      v2f a, b;
      a.x = Gb[kb]     * k[px * 32 + kb];
      a.y = Gb[kb + 1] * k[px * 32 + kb + 1];
      int nn = nt * 16 + mrow;
      b.x = Wp[kb * CDIM + nn];
      b.y = Wp[(kb + 1) * CDIM + nn];
      acc = wmma4(a, b, acc);
    }
#pragma unroll
    for (int r = 0; r < 8; ++r) {
      int m = r + khalf * 8;
      int n = nt * 16 + mrow;
      long p2 = pxbase + pxt * 16 + m;
      u[pxt * 16 + m][n] = acc[r] + pb[n] + q[p2 * 32 + n];
    }
  }
  __syncthreads();
  {
    v8f acc = {};
#pragma unroll
    for (int ks = 0; ks < 8; ++ks) {
      int kb = ks * 4 + khalf * 2;
      v2f a, b;
      a.x = u[pxt * 16 + mrow][kb];
      a.y = u[pxt * 16 + mrow][kb + 1];
      int nn = nt * 16 + mrow;
      b.x = Wf[kb * CDIM + nn];
      b.y = Wf[(kb + 1) * CDIM + nn];
      acc = wmma4(a, b, acc);
    }
#pragma unroll
    for (int r = 0; r < 8; ++r) {
      int m  = r + khalf * 8;
      int cc = nt * 16 + mrow;
      long p2 = pxbase + pxt * 16 + m;
      long sp = p2 % HW;
      long idx = ((long)bimg * CDIM + cc) * (long)HW + sp;
      out[idx] = res[idx] + ls1[cc] * (acc[r] + fb[cc]);
    }
  }
}

// ---------------------------------------------------------------------------
// Embedding: 3x3 stride-2 conv 32->64 (+bias) + BN, via im2col WMMA GEMM
// (K = 9 taps x 32 ch). Weights OIHW [64][32][3][3].
// ---------------------------------------------------------------------------
__global__ __launch_bounds__(256) void embed_kernel(
    const float* __restrict__ in, const float* __restrict__ w,
    const float* __restrict__ bias, const float* __restrict__ g,
    const float* __restrict__ be, const float* __restrict__ mu,
    const float* __restrict__ var, float* __restrict__ out) {
  const int lane  = threadIdx.x & 31;
  const int wave  = threadIdx.x >> 5;
  const int mrow  = lane & 15;
  const int khalf = lane >> 4;
  const long pxbase = (long)blockIdx.x * 64;

  for (int t = wave; t < 16; t += 8) {
    int pxt = t & 3, nt = t >> 2;
    long px = pxbase + pxt * 16 + mrow;
    int  n  = (int)(px / OHW);
    int  spo = (int)(px % OHW);
    int  oy = spo >> 7, ox = spo & 127;
    v8f acc = {};
#pragma unroll
    for (int tap = 0; tap < 9; ++tap) {
      int ty = tap / 3, tx = tap % 3;
      int iy = oy * 2 - 1 + ty;
      int ix = ox * 2 - 1 + tx;
      bool ok = ((unsigned)iy < (unsigned)HSZ) && ((unsigned)ix < (unsigned)WSZ);
      const float* ab = in + ((long)n * CDIM) * (long)HW + (long)iy * WSZ + ix;
#pragma unroll
      for (int ks = 0; ks < 8; ++ks) {
        int kb = ks * 4 + khalf * 2;
        v2f a, b;
        a.x = ok ? ab[(long)kb * HW] : 0.0f;
        a.y = ok ? ab[(long)(kb + 1) * HW] : 0.0f;
        int nn = nt * 16 + mrow;
        b.x = w[(nn * CDIM + kb) * 9 + tap];
        b.y = w[(nn * CDIM + kb + 1) * 9 + tap];
        acc = wmma4(a, b, acc);
      }
    }
#pragma unroll
    for (int r = 0; r < 8; ++r) {
      int m  = r + khalf * 8;
      int cc = nt * 16 + mrow;
      long p2 = pxbase + pxt * 16 + m;
      int  n2 = (int)(p2 / OHW);
      int  spo2 = (int)(p2 % OHW);
      float s = g[cc] * rsqrtf(var[cc] + 1e-5f);
      out[((long)n2 * CE + cc) * (long)OHW + spo2] =
          (acc[r] + bias[cc] - mu[cc]) * s + be[cc];
    }
  }
}

// ---------------------------------------------------------------------------
extern "C" void kernel_launch(void* const* d_in, const int* in_sizes, int n_in,
                              void* d_out, int out_size, void* d_ws,
                              size_t ws_size, hipStream_t stream) {
  (void)in_sizes; (void)n_in; (void)out_size; (void)ws_size;
#define IN(i) ((const float*)d_in[i])
  const float* x = IN(0);
  // encoder1
  const float *e_dw_w = IN(1), *e_dw_b = IN(2), *e_g = IN(3), *e_be = IN(4),
              *e_m = IN(5), *e_v = IN(6), *e_p1w = IN(7), *e_p1b = IN(8),
              *e_p2w = IN(9), *e_p2b = IN(10), *e_ls = IN(11);
  // local representation
  const float *l_dw_w = IN(12), *l_dw_b = IN(13), *l_g = IN(14), *l_be = IN(15),
              *l_m = IN(16), *l_v = IN(17), *l_p1w = IN(18), *l_p1b = IN(19),
              *l_p2w = IN(20), *l_p2b = IN(21), *l_ls = IN(22);
  // attention
  const float *a_qw = IN(23), *a_qb = IN(24), *a_kw = IN(25), *a_kb = IN(26),
              *a_wg = IN(27), *a_pw = IN(28), *a_pb = IN(29), *a_fw = IN(30),
              *a_fb = IN(31);
  const float *ls1 = IN(32), *ls2 = IN(33);
  // mlp
  const float *m_g = IN(34), *m_be = IN(35), *m_m = IN(36), *m_v = IN(37),
              *m_f1w = IN(38), *m_f1b = IN(39), *m_f2w = IN(40), *m_f2b = IN(41);
  // embedding
  const float *em_w = IN(42), *em_b = IN(43), *em_g = IN(44), *em_be = IN(45),
              *em_m = IN(46), *em_v = IN(47);
#undef IN

  float* ws   = (float*)d_ws;
  float* tmp  = ws;              // dw/bn scratch; later reused as q (token-major)
  float* hA   = tmp + P32;
  float* hB   = hA + P32;
  float* kbuf = hB + P32;        // k (token-major); later reused as BN scratch
  float* sums = kbuf + P32;      // [B][33] atomic partials
  float* G    = sums + BATCH * 33;

  const int elem_blocks = (int)(P32 / 256);          // 65536
  const int px_blocks   = (BATCH * HW) / 64;         // 8192
  const int emb_blocks  = (BATCH * OHW) / 64;        // 2048
  dim3 blk(256);

  // encoder1
  dwbn_kernel<<<elem_blocks, blk, 0, stream>>>(x, e_dw_w, e_dw_b, e_g, e_be,
                                               e_m, e_v, tmp);
  pw_block_kernel<128><<<px_blocks, blk, 0, stream>>>(x, tmp, e_p1w, e_p1b,
                                                      e_p2w, e_p2b, e_ls, hA);
  // local representation
  dwbn_kernel<<<elem_blocks, blk, 0, stream>>>(hA, l_dw_w, l_dw_b, l_g, l_be,
                                               l_m, l_v, tmp);
  pw_block_kernel<32><<<px_blocks, blk, 0, stream>>>(hA, tmp, l_p1w, l_p1b,
                                                     l_p2w, l_p2b, l_ls, hB);
  // attention
  hipMemsetAsync(sums, 0, BATCH * 33 * sizeof(float), stream);
  attn_qk_kernel<<<px_blocks, blk, 0, stream>>>(hB, a_qw, a_qb, a_kw, a_kb,
                                                a_wg, tmp /*q*/, kbuf /*k*/,
                                                sums);
  attn_finalize_kernel<<<1, 256, 0, stream>>>(sums, G);
  attn_out_kernel<<<px_blocks, blk, 0, stream>>>(hB, tmp, kbuf, G, a_pw, a_pb,
                                                 a_fw, a_fb, ls1, hA);
  // mlp
  bn_kernel<<<elem_blocks, blk, 0, stream>>>(hA, m_g, m_be, m_m, m_v, kbuf);
  pw_block_kernel<128><<<px_blocks, blk, 0, stream>>>(hA, kbuf, m_f1w, m_f1b,
                                                      m_f2w, m_f2b, ls2, hB);
  // downsampling embedding
  embed_kernel<<<emb_blocks, blk, 0, stream>>>(hB, em_w, em_b, em_g, em_be,
                                               em_m, em_v, (float*)d_out);
}